// QueryMambaOp_36206574305382
// MI455X (gfx1250) — compile-verified
//
#include <hip/hip_runtime.h>
#include <hip/hip_bf16.h>
#include <math.h>

// ---------------------------------------------------------------------------
// Problem constants (from the reference)
// ---------------------------------------------------------------------------
#define D_MODEL 768
#define D_STATE 16
#define D_CONV  4
#define D_INNER 1536          // 2 * D_MODEL
#define DT_RANK 48
#define BATCH   2
#define SEQLEN  2048
#define BL      (BATCH * SEQLEN)   // 4096 rows for all GEMMs

typedef __attribute__((ext_vector_type(2))) float v2f;
typedef __attribute__((ext_vector_type(8))) float v8f;

// ---------------------------------------------------------------------------
// fp32 GEMM on the CDNA5 fp32 matrix core (V_WMMA_F32_16X16X4_F32):
//   C[M x N] = act( A[M x K](row-major, lda) * W[N x K]^T (row-major, ldw) + bias[N] )
//
// Register blocking: each wave computes a 2(M) x 4(N) grid of 16x16 tiles
// (32x64 output), so per K-step we load 2 A-frags + 4 B-frags (6 x b64/lane)
// and issue 8 WMMAs -> 0.75 loads per WMMA.  Block = 256 threads = 8 waves
// arranged 2(M) x 4(N) -> block tile 64(M) x 256(N).
//
// EXEC stays all-ones around WMMA: whole-wave tiles outside N exit via a
// wave-uniform branch; partial tiles clamp load addresses per-lane and
// guard only the stores.  Prefetch is speculative (ISA 10.5) -> no guard.
// act: 0 = identity, 1 = softplus
// ---------------------------------------------------------------------------
__global__ __launch_bounds__(256)
void gemm_f32_wmma(const float* __restrict__ A, int lda,
                   const float* __restrict__ W, int ldw,
                   const float* __restrict__ bias,
                   float* __restrict__ C, int ldc,
                   int M, int N, int K, int act)
{
    const int lane   = threadIdx.x & 31;
    const int wave   = threadIdx.x >> 5;
    const int tileM  = blockIdx.y * 64  + (wave >> 2) * 32;
    const int tileN  = blockIdx.x * 256 + (wave & 3) * 64;
    if (tileN >= N) return;                 // wave-uniform exit

    const int half   = lane >> 4;           // 0: K={0,1}/rows 0-7 ; 1: K={2,3}/rows 8-15
    const int lane16 = lane & 15;

    const int m0 = tileM + lane16;          // A rows (always in range: M % 64 == 0)
    const int m1 = tileM + 16 + lane16;

    int   ncol[4];
    int   nclamp[4];
#pragma unroll
    for (int j = 0; j < 4; ++j) {
        ncol[j]   = tileN + 16 * j + lane16;
        nclamp[j] = (ncol[j] < N) ? ncol[j] : (N - 1);
    }

    const float* __restrict__ pA0 = A + (size_t)m0 * (size_t)lda;
    const float* __restrict__ pA1 = A + (size_t)m1 * (size_t)lda;
    const float* __restrict__ pB0 = W + (size_t)nclamp[0] * (size_t)ldw;
    const float* __restrict__ pB1 = W + (size_t)nclamp[1] * (size_t)ldw;
    const float* __restrict__ pB2 = W + (size_t)nclamp[2] * (size_t)ldw;
    const float* __restrict__ pB3 = W + (size_t)nclamp[3] * (size_t)ldw;

    v8f z8 = {0.f,0.f,0.f,0.f,0.f,0.f,0.f,0.f};
    v8f acc00 = z8, acc01 = z8, acc02 = z8, acc03 = z8;
    v8f acc10 = z8, acc11 = z8, acc12 = z8, acc13 = z8;

#pragma unroll 2
    for (int k = 0; k < K; k += 4) {
        const int k0 = k + half * 2;
        // speculative prefetch of next K-chunk (global_prefetch_b8, WGP scope)
        __builtin_prefetch(pA0 + k0 + 64, 0, 3);
        __builtin_prefetch(pB0 + k0 + 64, 0, 3);
        v2f a0 = *(const v2f*)(pA0 + k0);
        v2f a1 = *(const v2f*)(pA1 + k0);
        v2f b0 = *(const v2f*)(pB0 + k0);
        v2f b1 = *(const v2f*)(pB1 + k0);
        v2f b2 = *(const v2f*)(pB2 + k0);
        v2f b3 = *(const v2f*)(pB3 + k0);
        acc00 = __builtin_amdgcn_wmma_f32_16x16x4_f32(false, a0, false, b0, (short)0, acc00, false, false);
        acc01 = __builtin_amdgcn_wmma_f32_16x16x4_f32(false, a0, false, b1, (short)0, acc01, false, false);
        acc02 = __builtin_amdgcn_wmma_f32_16x16x4_f32(false, a0, false, b2, (short)0, acc02, false, false);
        acc03 = __builtin_amdgcn_wmma_f32_16x16x4_f32(false, a0, false, b3, (short)0, acc03, false, false);
        acc10 = __builtin_amdgcn_wmma_f32_16x16x4_f32(false, a1, false, b0, (short)0, acc10, false, false);
        acc11 = __builtin_amdgcn_wmma_f32_16x16x4_f32(false, a1, false, b1, (short)0, acc11, false, false);
        acc12 = __builtin_amdgcn_wmma_f32_16x16x4_f32(false, a1, false, b2, (short)0, acc12, false, false);
        acc13 = __builtin_amdgcn_wmma_f32_16x16x4_f32(false, a1, false, b3, (short)0, acc13, false, false);
    }

    float bv[4];
#pragma unroll
    for (int j = 0; j < 4; ++j) bv[j] = bias ? bias[nclamp[j]] : 0.0f;

    v8f* accr0[4] = { &acc00, &acc01, &acc02, &acc03 };
    v8f* accr1[4] = { &acc10, &acc11, &acc12, &acc13 };

    // D layout: lanes 0-15 hold rows sub+0..7 (vgpr v), lanes 16-31 rows sub+8..15
#pragma unroll
    for (int j = 0; j < 4; ++j) {
        if (ncol[j] >= N) continue;     // store guard only (no WMMA after this)
#pragma unroll
        for (int v = 0; v < 8; ++v) {
            const int mr0 = tileM + half * 8 + v;
            const int mr1 = tileM + 16 + half * 8 + v;
            float r0 = (*accr0[j])[v] + bv[j];
            float r1 = (*accr1[j])[v] + bv[j];
            if (act == 1) {                          // softplus, overflow-safe
                r0 = (r0 > 20.0f) ? r0 : __logf(1.0f + __expf(r0));
                r1 = (r1 > 20.0f) ? r1 : __logf(1.0f + __expf(r1));
            }
            C[(size_t)mr0 * (size_t)ldc + ncol[j]] = r0;
            C[(size_t)mr1 * (size_t)ldc + ncol[j]] = r1;
        }
    }
}

// ---------------------------------------------------------------------------
// Causal depthwise conv1d (width 4) + SiLU.  x = xz[..., :D_INNER]
// ---------------------------------------------------------------------------
__global__ __launch_bounds__(256)
void conv_silu_kernel(const float* __restrict__ xz, const float* __restrict__ conv_w,
                      const float* __restrict__ conv_b, float* __restrict__ x_act)
{
    const int idx = blockIdx.x * 256 + threadIdx.x;
    if (idx >= BL * D_INNER) return;
    const int d = idx % D_INNER;
    const int t = (idx / D_INNER) % SEQLEN;
    const int b = idx / (D_INNER * SEQLEN);

    float acc = conv_b[d];
#pragma unroll
    for (int k = 0; k < D_CONV; ++k) {
        const int l = t - (D_CONV - 1) + k;
        if (l >= 0) {
            acc += conv_w[d * D_CONV + k] *
                   xz[((size_t)(b * SEQLEN + l)) * (2 * D_INNER) + d];
        }
    }
    // silu
    x_act[idx] = acc / (1.0f + __expf(-acc));
}

// ---------------------------------------------------------------------------
// Selective scan. One thread per (b, d); 16 states live in registers.
// B/C vectors (uniform per block per timestep) are staged through LDS.
// 1536 % 256 == 0 so every block sees a single batch index b.
// ---------------------------------------------------------------------------
__global__ __launch_bounds__(256)
void scan_kernel(const float* __restrict__ dtb, const float* __restrict__ x_act,
                 const float* __restrict__ x_dbl, const float* __restrict__ A_log,
                 float* __restrict__ y)
{
    const int tid = blockIdx.x * 256 + threadIdx.x;   // 0 .. 3071
    const int b = tid / D_INNER;
    const int d = tid % D_INNER;

    float Acoef[D_STATE];
#pragma unroll
    for (int nn = 0; nn < D_STATE; ++nn)
        Acoef[nn] = -__expf(A_log[d * D_STATE + nn]);

    float h[D_STATE];
#pragma unroll
    for (int nn = 0; nn < D_STATE; ++nn) h[nn] = 0.0f;

    __shared__ float sBC[2 * D_STATE];                 // B[16] then C[16]

    for (int l = 0; l < SEQLEN; ++l) {
        const size_t row = (size_t)(b * SEQLEN + l);
        __syncthreads();
        if (threadIdx.x < 2 * D_STATE)
            sBC[threadIdx.x] = x_dbl[row * (DT_RANK + 2 * D_STATE) + DT_RANK + threadIdx.x];
        __syncthreads();

        const float dtv = dtb[row * D_INNER + d];
        const float xv  = x_act[row * D_INNER + d];
        const float dtx = dtv * xv;
        float yv = 0.0f;
#pragma unroll
        for (int nn = 0; nn < D_STATE; ++nn) {
            h[nn] = __expf(dtv * Acoef[nn]) * h[nn] + dtx * sBC[nn];
            yv += h[nn] * sBC[D_STATE + nn];
        }
        y[row * D_INNER + d] = yv;
    }
}

// ---------------------------------------------------------------------------
// Gating: y = (y + x*D) * silu(z) * silu(q)   (in place on y)
// ---------------------------------------------------------------------------
__global__ __launch_bounds__(256)
void gate_kernel(float* __restrict__ y, const float* __restrict__ x_act,
                 const float* __restrict__ xz, const float* __restrict__ q,
                 const float* __restrict__ Dp)
{
    const int idx = blockIdx.x * 256 + threadIdx.x;
    if (idx >= BL * D_INNER) return;
    const int d   = idx % D_INNER;
    const int row = idx / D_INNER;

    const float xv = x_act[idx];
    const float zv = xz[(size_t)row * (2 * D_INNER) + D_INNER + d];
    const float qv = q[idx];

    float r = y[idx] + xv * Dp[d];
    r *= zv / (1.0f + __expf(-zv));      // silu(z)
    r *= qv / (1.0f + __expf(-qv));      // silu(q)
    y[idx] = r;
}

// ---------------------------------------------------------------------------
// Host-side launcher
// ---------------------------------------------------------------------------
extern "C" void kernel_launch(void* const* d_in, const int* in_sizes, int n_in,
                              void* d_out, int out_size, void* d_ws, size_t ws_size,
                              hipStream_t stream)
{
    (void)in_sizes; (void)n_in; (void)out_size; (void)ws_size;

    const float* hidden    = (const float*)d_in[0];   // (B, L, 768)
    const float* query     = (const float*)d_in[1];   // (B, L, 768)
    const float* in_proj_w = (const float*)d_in[2];   // (3072, 768)
    const float* conv_w    = (const float*)d_in[3];   // (1536, 4)
    const float* conv_b    = (const float*)d_in[4];   // (1536,)
    const float* x_proj_w  = (const float*)d_in[5];   // (80, 1536)
    const float* dt_proj_w = (const float*)d_in[6];   // (1536, 48)
    const float* dt_proj_b = (const float*)d_in[7];   // (1536,)
    const float* A_log     = (const float*)d_in[8];   // (1536, 16)
    const float* Dp        = (const float*)d_in[9];   // (1536,)
    const float* query_w   = (const float*)d_in[10];  // (1536, 768)
    const float* query_b   = (const float*)d_in[11];  // (1536,)
    const float* out_w     = (const float*)d_in[12];  // (768, 1536)
    float* out = (float*)d_out;                       // (B, L, 768)

    // Workspace carving (floats)
    float* ws   = (float*)d_ws;
    float* xz    = ws;                                 // BL * 3072
    float* x_act = xz    + (size_t)BL * (2 * D_INNER); // BL * 1536
    float* x_dbl = x_act + (size_t)BL * D_INNER;       // BL * 80
    float* dtb   = x_dbl + (size_t)BL * (DT_RANK + 2 * D_STATE); // BL * 1536
    float* qb    = dtb   + (size_t)BL * D_INNER;       // BL * 1536
    float* yb    = qb    + (size_t)BL * D_INNER;       // BL * 1536

    dim3 blk(256);
    // Block tile = 64(M) x 256(N); M = 4096 is a multiple of 64.
    #define GEMM_GRID(Ncols) dim3(((Ncols) + 255) / 256, BL / 64)

    // 1) in_proj: xz = hidden @ in_proj_w^T        (4096 x 3072, K=768)
    gemm_f32_wmma<<<GEMM_GRID(2 * D_INNER), blk, 0, stream>>>(
        hidden, D_MODEL, in_proj_w, D_MODEL, nullptr,
        xz, 2 * D_INNER, BL, 2 * D_INNER, D_MODEL, 0);

    // 2) causal depthwise conv + silu -> x_act
    conv_silu_kernel<<<(BL * D_INNER + 255) / 256, blk, 0, stream>>>(
        xz, conv_w, conv_b, x_act);

    // 3) x_proj: x_dbl = x_act @ x_proj_w^T        (4096 x 80, K=1536)
    gemm_f32_wmma<<<GEMM_GRID(DT_RANK + 2 * D_STATE), blk, 0, stream>>>(
        x_act, D_INNER, x_proj_w, D_INNER, nullptr,
        x_dbl, DT_RANK + 2 * D_STATE, BL, DT_RANK + 2 * D_STATE, D_INNER, 0);

    // 4) dt_proj + softplus: dt = softplus(x_dbl[:, :48] @ dt_proj_w^T + b)  (4096 x 1536, K=48)
    gemm_f32_wmma<<<GEMM_GRID(D_INNER), blk, 0, stream>>>(
        x_dbl, DT_RANK + 2 * D_STATE, dt_proj_w, DT_RANK, dt_proj_b,
        dtb, D_INNER, BL, D_INNER, DT_RANK, 1);

    // 5) query proj: q = query @ query_w^T + query_b   (4096 x 1536, K=768)
    gemm_f32_wmma<<<GEMM_GRID(D_INNER), blk, 0, stream>>>(
        query, D_MODEL, query_w, D_MODEL, query_b,
        qb, D_INNER, BL, D_INNER, D_MODEL, 0);

    // 6) selective scan -> y
    scan_kernel<<<(BATCH * D_INNER) / 256, blk, 0, stream>>>(
        dtb, x_act, x_dbl, A_log, yb);

    // 7) gating (skip via D, silu(z), silu(q)) in place on y
    gate_kernel<<<(BL * D_INNER + 255) / 256, blk, 0, stream>>>(
        yb, x_act, xz, qb, Dp);

    // 8) out_proj: out = y @ out_proj_w^T          (4096 x 768, K=1536)
    gemm_f32_wmma<<<GEMM_GRID(D_MODEL), blk, 0, stream>>>(
        yb, D_INNER, out_w, D_INNER, nullptr,
        out, D_MODEL, BL, D_MODEL, D_INNER, 0);
}